// LLObjectCondensation_36197984371049
// MI455X (gfx1250) — compile-verified
//
#include <hip/hip_runtime.h>
#include <hip/hip_bf16.h>

#define KMAX 128
#define BB   8

typedef float v2f __attribute__((ext_vector_type(2)));
typedef float v8f __attribute__((ext_vector_type(8)));

// ---------------- workspace layout (bytes) ----------------
// [    0,  8192)  unsigned long long cand[B*K]   (argmax keys)
// [ 8192, 12288)  float oxax[B*K]
// [12288, 16384)  float oxay[B*K]
// [16384, 20480)  float oqa [B*K]
// [20480, 24576)  float oxa2[B*K]
// [24576, 28672)  float opres[B*K]
// [28672, 28800)  float attr[8], rep[8], lbeta[8], noise[8]
// [28800, 28864)  int   nobj[8], nnoise[8]
#define WS_WORDS 7216   // 28864/4

__device__ __forceinline__ float qfun(float bc) {
    // arctanh(bc/1.002)^2 + 0.5  (per-point, not per-pair: full-precision log is fine)
    float x  = bc * (1.0f / 1.002f);
    float at = 0.5f * logf((1.0f + x) / (1.0f - x));
    return at * at + 0.5f;
}

__device__ __forceinline__ float clipb(float b) {
    return fminf(fmaxf(b, 1e-6f), 1.0f - 1e-5f);
}

__global__ void zero_ws_kernel(unsigned int* w, int nwords) {
    int i = blockIdx.x * blockDim.x + threadIdx.x;
    if (i < nwords) w[i] = 0u;
}

// Phase 1: argmax candidates per (b,k) + noise accumulation.
__global__ void scan_kernel(const float* __restrict__ beta,
                            const int* __restrict__ t_idx,
                            unsigned long long* __restrict__ cand,
                            float* __restrict__ noise_sum,
                            int* __restrict__ nnoise, int N) {
    int idx = blockIdx.x * blockDim.x + threadIdx.x;
    if (idx >= BB * N) return;
    int b = idx / N;
    int n = idx - b * N;
    float bc = clipb(beta[idx]);
    int  k  = t_idx[idx];
    unsigned int bits = __float_as_uint(bc);   // bc > 0: bit pattern is order-preserving
    unsigned long long key =
        ((unsigned long long)bits << 32) | (unsigned long long)(0xFFFFFFFFu - (unsigned)n);
    atomicMax(&cand[b * KMAX + k], key);
    if (k == 0) {
        atomicAdd(&noise_sum[b], bc);
        atomicAdd(&nnoise[b], 1);
    }
}

// Phase 2: decode condensation points, build per-(b,k) tables, L_beta terms.
__global__ void finalize_alpha_kernel(const float* __restrict__ ccoords,
                                      const unsigned long long* __restrict__ cand,
                                      float* __restrict__ oxax, float* __restrict__ oxay,
                                      float* __restrict__ oqa,  float* __restrict__ oxa2,
                                      float* __restrict__ opres,
                                      float* __restrict__ lbeta, int* __restrict__ nobj,
                                      int N) {
    int tid = blockIdx.x * blockDim.x + threadIdx.x;
    if (tid >= BB * KMAX) return;
    int b = tid / KMAX;
    int k = tid - b * KMAX;
    unsigned long long c = cand[tid];
    unsigned int hi = (unsigned int)(c >> 32);
    bool present = (k > 0) && (hi != 0u);
    float xax = 0.f, xay = 0.f, qa = 0.f, xa2 = 0.f;
    if (present) {
        unsigned int n = 0xFFFFFFFFu - (unsigned int)(c & 0xFFFFFFFFull);
        float ba = __uint_as_float(hi);
        size_t g = ((size_t)b * N + n) * 2;
        xax = ccoords[g + 0];
        xay = ccoords[g + 1];
        qa  = qfun(ba);
        xa2 = xax * xax + xay * xay;
        atomicAdd(&lbeta[b], 1.0f - ba);
        atomicAdd(&nobj[b], 1);
    }
    oxax[tid] = xax; oxay[tid] = xay; oqa[tid] = qa; oxa2[tid] = xa2;
    opres[tid] = present ? 1.0f : 0.0f;
}

// Phase 3: N x K pairwise potential via V_WMMA_F32_16X16X4_F32 (C=2 padded to K=4).
__global__ void __launch_bounds__(256)
pairwise_kernel(const float* __restrict__ ccoords,
                const float* __restrict__ beta,
                const int* __restrict__ t_idx,
                const float* __restrict__ oxax, const float* __restrict__ oxay,
                const float* __restrict__ oqa,  const float* __restrict__ oxa2,
                const float* __restrict__ opres,
                float* __restrict__ attr_acc, float* __restrict__ rep_acc, int N) {
    __shared__ float s_xax[KMAX], s_xay[KMAX], s_qa[KMAX], s_xa2[KMAX], s_pres[KMAX];
    __shared__ float s_xn2[8][16], s_q[8][16];
    __shared__ int   s_t[8][16];

    const int b    = blockIdx.y;
    const int lane = threadIdx.x;              // 0..31 (wave32)
    const int wy   = threadIdx.y;              // 0..7  (8 waves / block)
    const int tid  = wy * 32 + lane;
    const int ntiles = (N + 15) >> 4;

    // Cooperative load of condensation-point tables for this batch.
    for (int i = tid; i < KMAX; i += 256) {
        int g = b * KMAX + i;
        s_xax[i]  = oxax[g];  s_xay[i] = oxay[g];
        s_qa[i]   = oqa[g];   s_xa2[i] = oxa2[g];
        s_pres[i] = opres[g];
    }

    const int  tile  = blockIdx.x * 8 + wy;    // wave-uniform
    const bool valid = (tile < ntiles);        // wave-uniform -> EXEC full at WMMA

    // A matrix (16x4 f32): lanes 0-15 carry K=0,1 = (x,y); lanes 16-31 carry K=2,3 = 0.
    v2f a; a.x = 0.f; a.y = 0.f;
    if (lane < 16) {
        int n = tile * 16 + lane;
        if (valid && n < N) {
            size_t g = (size_t)b * N + n;
            float x = ccoords[g * 2 + 0];
            float y = ccoords[g * 2 + 1];
            a.x = x; a.y = y;
            s_xn2[wy][lane] = x * x + y * y;
            s_q[wy][lane]   = qfun(clipb(beta[g]));
            s_t[wy][lane]   = t_idx[g];
        } else {
            s_xn2[wy][lane] = 0.f;
            s_q[wy][lane]   = 0.f;   // zero q kills contributions of padded rows
            s_t[wy][lane]   = -1;
        }
    }
    __syncthreads();

    float attr = 0.f, rep = 0.f;
    if (valid) {
        const int  mbase = (lane >> 4) * 8;    // C/D layout: VGPR r -> M=r (lanes 0-15), M=8+r
        const int  jj    = lane & 15;          // column within 16-wide k tile
        const bool lolane = (lane < 16);
        float xn2r[8], qr[8]; int tr[8];
        #pragma unroll
        for (int r = 0; r < 8; ++r) {
            xn2r[r] = s_xn2[wy][mbase + r];
            qr[r]   = s_q[wy][mbase + r];
            tr[r]   = s_t[wy][mbase + r];
        }
        #pragma unroll
        for (int kt = 0; kt < KMAX / 16; ++kt) {
            const int jg = kt * 16 + jj;
            // per-column epilogue constants (same for both lane halves)
            const float qaj  = s_qa[jg];
            const float xa2j = s_xa2[jg];
            const float paj  = s_pres[jg] * qaj;   // pres * q_alpha
            // B matrix (4x16 f32): lanes 0-15 carry K=0,1 rows; lanes 16-31 K=2,3 = 0.
            // Branchless masking: load for all lanes, cndmask-zero the upper half.
            float bx = s_xax[jg];
            float by = s_xay[jg];
            v2f bv;
            bv.x = lolane ? bx : 0.0f;
            bv.y = lolane ? by : 0.0f;
            v8f c = {};
            c = __builtin_amdgcn_wmma_f32_16x16x4_f32(
                    /*neg_a=*/false, a, /*neg_b=*/false, bv,
                    /*c_mod=*/(short)0, c, /*reuse_a=*/false, /*reuse_b=*/false);
            #pragma unroll
            for (int r = 0; r < 8; ++r) {
                float dot   = c[r];
                float d2    = fmaxf(xn2r[r] + xa2j - 2.0f * dot, 0.0f);
                float dd    = __builtin_amdgcn_sqrtf(d2 + 1e-6f);   // raw v_sqrt_f32
                float hinge = fmaxf(0.0f, 1.0f - dd);
                float pq    = paj * qr[r];                          // pres*q_alpha*q_n
                float isM   = (tr[r] == jg) ? 1.0f : 0.0f;          // cmp + cndmask, no branch
                attr = fmaf(isM * d2, pq, attr);
                rep  = fmaf((1.0f - isM) * hinge, pq, rep);
            }
        }
    }

    // Wave32 reduction, one atomic per wave per accumulator.
    #pragma unroll
    for (int off = 16; off > 0; off >>= 1) {
        attr += __shfl_xor(attr, off, 32);
        rep  += __shfl_xor(rep,  off, 32);
    }
    if (valid && lane == 0) {
        atomicAdd(&attr_acc[b], attr);
        atomicAdd(&rep_acc[b],  rep);
    }
}

// Phase 4: scalar combine.
__global__ void combine_kernel(const float* __restrict__ attr, const float* __restrict__ rep,
                               const float* __restrict__ lbeta, const float* __restrict__ noise,
                               const int* __restrict__ nobj, const int* __restrict__ nnoise,
                               float* __restrict__ out, int N) {
    if (threadIdx.x == 0 && blockIdx.x == 0) {
        float s = 0.f;
        for (int b = 0; b < BB; ++b) {
            float va = attr[b]  / (float)N;
            float vr = rep[b]   / (float)N;
            float lb = lbeta[b] / (float)(nobj[b]   > 1 ? nobj[b]   : 1);
            float ln = noise[b] / (float)(nnoise[b] > 1 ? nnoise[b] : 1);
            s += va + vr + lb + ln;
        }
        out[0] = s / (float)BB;
    }
}

extern "C" void kernel_launch(void* const* d_in, const int* in_sizes, int n_in,
                              void* d_out, int out_size, void* d_ws, size_t ws_size,
                              hipStream_t stream) {
    const float* ccoords = (const float*)d_in[0];
    const float* beta    = (const float*)d_in[1];
    const int*   t_idx   = (const int*)d_in[2];
    const int N = in_sizes[1] / BB;   // beta is (B, N)

    char* ws = (char*)d_ws;
    unsigned long long* cand = (unsigned long long*)ws;
    float* oxax  = (float*)(ws + 8192);
    float* oxay  = (float*)(ws + 12288);
    float* oqa   = (float*)(ws + 16384);
    float* oxa2  = (float*)(ws + 20480);
    float* opres = (float*)(ws + 24576);
    float* attr  = (float*)(ws + 28672);
    float* rep   = attr + 8;
    float* lbeta = attr + 16;
    float* noise = attr + 24;
    int*   nobj   = (int*)(attr + 32);
    int*   nnoise = nobj + 8;

    zero_ws_kernel<<<(WS_WORDS + 255) / 256, 256, 0, stream>>>((unsigned int*)ws, WS_WORDS);

    const int total = BB * N;
    scan_kernel<<<(total + 255) / 256, 256, 0, stream>>>(beta, t_idx, cand, noise, nnoise, N);

    finalize_alpha_kernel<<<(BB * KMAX + 255) / 256, 256, 0, stream>>>(
        ccoords, cand, oxax, oxay, oqa, oxa2, opres, lbeta, nobj, N);

    const int ntiles = (N + 15) / 16;          // 2500 for N=40000
    dim3 grid((ntiles + 7) / 8, BB);
    dim3 block(32, 8);                         // 8 wave32s per block
    pairwise_kernel<<<grid, block, 0, stream>>>(
        ccoords, beta, t_idx, oxax, oxay, oqa, oxa2, opres, attr, rep, N);

    combine_kernel<<<1, 1, 0, stream>>>(attr, rep, lbeta, noise, nobj, nnoise, (float*)d_out, N);
}